// Top2Gate_62474594288231
// MI455X (gfx1250) — compile-verified
//
#include <hip/hip_runtime.h>
#include <hip/hip_bf16.h>

typedef __attribute__((ext_vector_type(2))) float v2f;
typedef __attribute__((ext_vector_type(8))) float v8f;

#define D_DIM 1024
#define E_DIM 8
#define WSTRIDE 1028            // pad rows: 4112 B = 257*16 -> bank-spread + 16B aligned
#define TOK_PER_WAVE 16
#define WAVES_PER_BLOCK 8
#define TOK_PER_BLOCK (TOK_PER_WAVE * WAVES_PER_BLOCK)
#define GATE_EPS 1e-9f

// Generic -> LDS(addrspace 3) pointer value == 32-bit LDS byte offset on gfx1250.
__device__ __forceinline__ unsigned lds_byte_off(const void* p) {
    return (unsigned)(unsigned long long)
        (const __attribute__((address_space(3))) void*)p;
}

__global__ __launch_bounds__(32) void init_ws_kernel(float* ws) {
    if (threadIdx.x < 16) ws[threadIdx.x] = 0.0f;
}

__global__ __launch_bounds__(256) void top2gate_kernel(
        const float* __restrict__ x,      // [tokens, D]
        const float* __restrict__ Wm,     // [E, D]
        const float* __restrict__ noise,  // [tokens, E]
        float* __restrict__ out,          // [tokens, E+1] dispatch mask
        float* __restrict__ ws,           // [16] global me/ce accumulators
        int tokens) {
    // Row E_DIM (index 8) is an all-zero row: lanes n>=8 read it -> branch-free B frags.
    __shared__ float Wl[(E_DIM + 1) * WSTRIDE];
    __shared__ float Cl[WAVES_PER_BLOCK][TOK_PER_WAVE * 16]; // per-wave C-tile transpose
    __shared__ float red[16];                                // block partial me[8] / ce[8]

    const int tid  = threadIdx.x;
    const int w    = tid >> 5;   // wave in block
    const int lane = tid & 31;
    const int n    = lane & 15;  // column (expert, padded to 16)
    const int h    = lane >> 4;  // half-wave: K offset selector

    // --- Stage W into LDS with async global->LDS b128 copies (ASYNCcnt path). ---
    for (int i = tid; i < (E_DIM * D_DIM) / 4; i += 256) {
        const int idx = i * 4;                    // float index, 16B granules
        const int r = idx >> 10, c = idx & (D_DIM - 1);
        const unsigned dst = lds_byte_off(&Wl[r * WSTRIDE + c]);
        const float* src = Wm + idx;
        asm volatile("global_load_async_to_lds_b128 %0, %1, off"
                     :: "v"(dst), "v"(src) : "memory");
    }
    // Zero the padding row (read by lanes n>=8) and the block reducers.
    for (int i = tid; i < D_DIM; i += 256) Wl[E_DIM * WSTRIDE + i] = 0.0f;
    if (tid < 16) red[tid] = 0.0f;
    asm volatile("s_wait_asynccnt 0" ::: "memory");
    __syncthreads();

    const int token = blockIdx.x * TOK_PER_BLOCK + w * TOK_PER_WAVE + n; // A-row
    const float* __restrict__ xrow = x + (size_t)token * D_DIM;
    const float* __restrict__ wrow = Wl + (n < E_DIM ? n : E_DIM) * WSTRIDE;

    // D = A(16x4 f32) x B(4x16 f32) + C, K=1024 in steps of 4.
    v8f acc = {0.f, 0.f, 0.f, 0.f, 0.f, 0.f, 0.f, 0.f};
#pragma unroll 8
    for (int k = 0; k < D_DIM; k += 4) {
        // A fragment: lane holds row M=n, K = k + 2h + {0,1}
        v2f a = *(const v2f*)(xrow + k + 2 * h);
        // B fragment: lane holds col N=n, K = k + 2h + {0,1}; row 8 is zeros
        v2f b = *(const v2f*)(wrow + k + 2 * h);
        acc = __builtin_amdgcn_wmma_f32_16x16x4_f32(
            /*neg_a=*/false, a, /*neg_b=*/false, b,
            /*c_mod=*/(short)0, acc, /*reuse_a=*/false, /*reuse_b=*/false);
    }

    // Spill C tile: lane holds column n for rows M = j + 8h -> Cl[w][row*16 + col]
#pragma unroll
    for (int j = 0; j < 8; ++j)
        Cl[w][(j + 8 * h) * 16 + n] = acc[j];
    // Same-wave LDS ops are in-order; no barrier needed before same-wave readback.

    // Epilogue: lanes 0..15 each own one token of this wave's tile.
    if (lane < 16) {
        const int t = lane;
        const int mytok = blockIdx.x * TOK_PER_BLOCK + w * TOK_PER_WAVE + t;
        const float* __restrict__ nrow = noise + (size_t)mytok * E_DIM;

        float z[E_DIM];
        float zmax = -3.4e38f;
#pragma unroll
        for (int e = 0; e < E_DIM; ++e) {
            float logit = Cl[w][t * 16 + e];
            float u = nrow[e];
            float g = -logf(-logf(u + GATE_EPS) + GATE_EPS); // gumbel noise
            z[e] = logit + g;
            zmax = fmaxf(zmax, z[e]);
        }
        float s[E_DIM];
        float ssum = 0.0f;
#pragma unroll
        for (int e = 0; e < E_DIM; ++e) { s[e] = expf(z[e] - zmax); ssum += s[e]; }
        const float inv = 1.0f / ssum;
#pragma unroll
        for (int e = 0; e < E_DIM; ++e) {
            s[e] *= inv;
            atomicAdd(&red[e],     s[e]);          // Σ scores  (-> me)
            atomicAdd(&red[8 + e], s[e] * s[e]);   // Σ scores² (-> ce)
        }

        // top-2 (strict > keeps the lowest index on ties, matching lax.top_k)
        int i1 = 0; float s1 = s[0];
#pragma unroll
        for (int e = 1; e < E_DIM; ++e) if (s[e] > s1) { s1 = s[e]; i1 = e; }
        int i2 = -1; float s2 = -1.0f;
#pragma unroll
        for (int e = 0; e < E_DIM; ++e)
            if (e != i1 && s[e] > s2) { s2 = s[e]; i2 = e; }

        float* __restrict__ orow = out + (size_t)mytok * (E_DIM + 1);
        orow[0] = 1.0f; // shared expert slot
#pragma unroll
        for (int e = 0; e < E_DIM; ++e)
            orow[1 + e] = (e == i1) ? s1 : ((e == i2) ? s2 : 0.0f);
    }

    __syncthreads();
    if (tid < 16) atomicAdd(&ws[tid], red[tid]);
}

__global__ __launch_bounds__(32) void loss_kernel(const float* __restrict__ ws,
                                                  float* __restrict__ out,
                                                  int tokens, int out_idx) {
    if (threadIdx.x == 0) {
        const float invN = 1.0f / (float)tokens;
        float acc = 0.0f;
        for (int e = 0; e < E_DIM; ++e)
            acc += (ws[e] * invN) * (ws[8 + e] * invN);
        out[out_idx] = acc * (float)(E_DIM * E_DIM);
    }
}

extern "C" void kernel_launch(void* const* d_in, const int* in_sizes, int n_in,
                              void* d_out, int out_size, void* d_ws, size_t ws_size,
                              hipStream_t stream) {
    const float* x     = (const float*)d_in[0];  // [B,S,D] f32
    const float* Wm    = (const float*)d_in[1];  // [E,D]   f32
    const float* noise = (const float*)d_in[2];  // [B,S,E] f32
    float* out = (float*)d_out;
    float* ws  = (float*)d_ws;

    const int tokens = in_sizes[2] / E_DIM;      // B*S = 32768
    const int blocks = tokens / TOK_PER_BLOCK;   // 256

    init_ws_kernel<<<1, 32, 0, stream>>>(ws);
    top2gate_kernel<<<blocks, 256, 0, stream>>>(x, Wm, noise, out, ws, tokens);
    loss_kernel<<<1, 32, 0, stream>>>(ws, out, tokens, tokens * (E_DIM + 1));
}